// GAT_47459388620815
// MI455X (gfx1250) — compile-verified
//
#include <hip/hip_runtime.h>
#include <hip/hip_bf16.h>

typedef float v2f __attribute__((ext_vector_type(2)));
typedef float v8f __attribute__((ext_vector_type(8)));

#define HEADS 8
#define CH 16
#define F 128           // DIN == H*C == 128
#define NEG_SLOPE 0.2f

#if defined(__has_builtin)
#if __has_builtin(__builtin_amdgcn_wmma_f32_16x16x4_f32)
#define HAVE_WMMA_F32X4 1
#endif
#endif

__device__ __forceinline__ unsigned flipf(float f) {
  unsigned u = __float_as_uint(f);
  return (u >> 31) ? ~u : (u | 0x80000000u);
}
__device__ __forceinline__ float unflipf(unsigned u) {
  return __uint_as_float((u >> 31) ? (u ^ 0x80000000u) : ~u);
}
__device__ __forceinline__ float lrelu(float v) {
  return v > 0.0f ? v : NEG_SLOPE * v;
}

// ---------------------------------------------------------------------------
// 0) init: out = bias (broadcast), m = flip(-inf), denom = 0
// ---------------------------------------------------------------------------
__global__ __launch_bounds__(256) void gat_init(float* __restrict__ out,
                                                const float* __restrict__ bias,
                                                unsigned* __restrict__ m_u,
                                                float* __restrict__ denom,
                                                int N) {
  long long i = (long long)blockIdx.x * 256 + threadIdx.x;
  long long tot = (long long)N * F;
  if (i < tot) out[i] = bias[(int)(i & (F - 1))];
  if (i < (long long)N * HEADS) {
    m_u[i] = 0x007FFFFFu;   // flip(-inf)
    denom[i] = 0.0f;
  }
}

// ---------------------------------------------------------------------------
// 1) h = x @ W via V_WMMA_F32_16X16X4_F32, fused a_src/a_dst epilogue.
//    Block = 256 threads = 8 waves; wave w owns head w (cols 16w..16w+15);
//    block handles one 16-row strip of x staged in LDS.
// ---------------------------------------------------------------------------
__global__ __launch_bounds__(256) void gat_gemm_wmma(
    const float* __restrict__ x, const float* __restrict__ W,
    const float* __restrict__ att_src, const float* __restrict__ att_dst,
    float* __restrict__ h, float* __restrict__ a_src, float* __restrict__ a_dst,
    int N) {
  __shared__ float sA[16 * F];

  const int tid = threadIdx.x;
  const int wave = tid >> 5;        // 0..7 == head / 16-col tile
  const int lane = tid & 31;
  const int half = lane >> 4;       // 0: lanes 0-15, 1: lanes 16-31
  const int l16 = lane & 15;
  const int rowbase = blockIdx.x * 16;

  // Cooperative load of the 16x128 x-strip into LDS (row-clamped for tail).
  {
    int r = tid >> 4;                 // 0..15
    int c = (tid & 15) << 3;          // 0,8,...,120
    int rg = rowbase + r;
    if (rg >= N) rg = N - 1;
    const float4* s4 = (const float4*)(x + (size_t)rg * F + c);
    float4 a0 = s4[0];
    float4 a1 = s4[1];
    *(float4*)(sA + r * F + c) = a0;
    *(float4*)(sA + r * F + c + 4) = a1;
  }
  __syncthreads();

  const int colbase = wave * CH;
  v8f acc = {0.f, 0.f, 0.f, 0.f, 0.f, 0.f, 0.f, 0.f};

  for (int k = 0; k < F; k += 4) {
#if defined(HAVE_WMMA_F32X4)
    // A 16x4 f32 layout: lane(<16) holds A[l16][k+0..1], lane(>=16) A[l16][k+2..3]
    v2f afrag;
    const float* ap = sA + l16 * F + k + 2 * half;
    afrag.x = ap[0];
    afrag.y = ap[1];
    // B 4x16 f32 layout (mirrors A): lane holds W[k+2*half+{0,1}][colbase+l16]
    v2f bfrag;
    const float* bp = W + (size_t)(k + 2 * half) * F + colbase + l16;
    bfrag.x = bp[0];
    bfrag.y = bp[F];
    acc = __builtin_amdgcn_wmma_f32_16x16x4_f32(false, afrag, false, bfrag,
                                                (short)0, acc, false, false);
#else
    for (int v = 0; v < 8; ++v) {
      int row = v + 8 * half;
      for (int j = 0; j < 4; ++j)
        acc[v] += sA[row * F + k + j] * W[(size_t)(k + j) * F + colbase + l16];
    }
#endif
  }

  // Epilogue: store h tile; reduce per-row dots with att vectors.
  // D layout: VGPR v of lane -> M = v + 8*half, N(channel) = l16.
  const float was = att_src[wave * CH + l16];
  const float wad = att_dst[wave * CH + l16];
#pragma unroll
  for (int v = 0; v < 8; ++v) {
    int row = rowbase + v + 8 * half;
    float val = acc[v];
    float ps = val * was;
    float pd = val * wad;
#pragma unroll
    for (int off = 8; off >= 1; off >>= 1) {   // stays within 16-lane halves
      ps += __shfl_xor(ps, off, 32);
      pd += __shfl_xor(pd, off, 32);
    }
    if (row < N) {
      h[(size_t)row * F + colbase + l16] = val;
      if (l16 == 0) {
        a_src[(size_t)row * HEADS + wave] = ps;
        a_dst[(size_t)row * HEADS + wave] = pd;
      }
    }
  }
}

// ---------------------------------------------------------------------------
// 2) per-edge logits -> segment max via GLOBAL_ATOMIC_MAX_U32 (flipped floats)
// ---------------------------------------------------------------------------
__global__ __launch_bounds__(256) void gat_logits_max(
    const int* __restrict__ srcA, const int* __restrict__ dstA,
    const float* __restrict__ a_src, const float* __restrict__ a_dst,
    unsigned* __restrict__ m_u, int E, int EN) {
  int e = blockIdx.x * 256 + threadIdx.x;
  if (e >= EN) return;
  int s, d;
  if (e < E) { s = srcA[e]; d = dstA[e]; } else { s = d = e - E; }
  const float4* as4 = (const float4*)(a_src + (size_t)s * HEADS);
  const float4* ad4 = (const float4*)(a_dst + (size_t)d * HEADS);
  float4 s0 = as4[0], s1 = as4[1];
  float4 d0 = ad4[0], d1 = ad4[1];
  unsigned* mp = m_u + (size_t)d * HEADS;
  atomicMax(mp + 0, flipf(lrelu(s0.x + d0.x)));
  atomicMax(mp + 1, flipf(lrelu(s0.y + d0.y)));
  atomicMax(mp + 2, flipf(lrelu(s0.z + d0.z)));
  atomicMax(mp + 3, flipf(lrelu(s0.w + d0.w)));
  atomicMax(mp + 4, flipf(lrelu(s1.x + d1.x)));
  atomicMax(mp + 5, flipf(lrelu(s1.y + d1.y)));
  atomicMax(mp + 6, flipf(lrelu(s1.z + d1.z)));
  atomicMax(mp + 7, flipf(lrelu(s1.w + d1.w)));
}

// ---------------------------------------------------------------------------
// 3) unflip m in place (every node has a self loop => always finite)
// ---------------------------------------------------------------------------
__global__ __launch_bounds__(256) void gat_finalize_m(unsigned* __restrict__ m_u,
                                                      int count) {
  int i = blockIdx.x * 256 + threadIdx.x;
  if (i < count) {
    float v = unflipf(m_u[i]);
    ((float*)m_u)[i] = v;
  }
}

// ---------------------------------------------------------------------------
// 4) per-edge exp(e - m[dst]) -> segment sum via GLOBAL_ATOMIC_ADD_F32
// ---------------------------------------------------------------------------
__global__ __launch_bounds__(256) void gat_expsum(
    const int* __restrict__ srcA, const int* __restrict__ dstA,
    const float* __restrict__ a_src, const float* __restrict__ a_dst,
    const float* __restrict__ m, float* __restrict__ denom, int E, int EN) {
  int e = blockIdx.x * 256 + threadIdx.x;
  if (e >= EN) return;
  int s, d;
  if (e < E) { s = srcA[e]; d = dstA[e]; } else { s = d = e - E; }
  const float4* as4 = (const float4*)(a_src + (size_t)s * HEADS);
  const float4* ad4 = (const float4*)(a_dst + (size_t)d * HEADS);
  const float4* m4 = (const float4*)(m + (size_t)d * HEADS);
  float4 s0 = as4[0], s1 = as4[1];
  float4 d0 = ad4[0], d1 = ad4[1];
  float4 m0 = m4[0], m1 = m4[1];
  float* dp = denom + (size_t)d * HEADS;
  atomicAdd(dp + 0, expf(lrelu(s0.x + d0.x) - m0.x));
  atomicAdd(dp + 1, expf(lrelu(s0.y + d0.y) - m0.y));
  atomicAdd(dp + 2, expf(lrelu(s0.z + d0.z) - m0.z));
  atomicAdd(dp + 3, expf(lrelu(s0.w + d0.w) - m0.w));
  atomicAdd(dp + 4, expf(lrelu(s1.x + d1.x) - m1.x));
  atomicAdd(dp + 5, expf(lrelu(s1.y + d1.y) - m1.y));
  atomicAdd(dp + 6, expf(lrelu(s1.z + d1.z) - m1.z));
  atomicAdd(dp + 7, expf(lrelu(s1.w + d1.w) - m1.w));
}

// ---------------------------------------------------------------------------
// 5) aggregation: one wave per edge; lane owns 4 consecutive channels of the
//    128-wide feature (head = lane/4). Coalesced float4 gather of h[src],
//    4x GLOBAL_ATOMIC_ADD_F32 scatter into out[dst].
// ---------------------------------------------------------------------------
__global__ __launch_bounds__(256) void gat_aggregate(
    const int* __restrict__ srcA, const int* __restrict__ dstA,
    const float* __restrict__ a_src, const float* __restrict__ a_dst,
    const float* __restrict__ m, const float* __restrict__ denom,
    const float* __restrict__ h, float* __restrict__ out, int E, int EN) {
  long long widx = (long long)blockIdx.x * 8 + (threadIdx.x >> 5);
  if (widx >= EN) return;
  int e = (int)widx;
  int lane = threadIdx.x & 31;
  int s, d;
  if (e < E) { s = srcA[e]; d = dstA[e]; } else { s = d = e - E; }
  int hd = lane >> 2;   // head index for this lane's 4 channels
  float lg = lrelu(a_src[(size_t)s * HEADS + hd] + a_dst[(size_t)d * HEADS + hd]);
  float alpha = expf(lg - m[(size_t)d * HEADS + hd]) / denom[(size_t)d * HEADS + hd];
  float4 hv = *(const float4*)(h + (size_t)s * F + lane * 4);
  float* op = out + (size_t)d * F + lane * 4;
  atomicAdd(op + 0, alpha * hv.x);
  atomicAdd(op + 1, alpha * hv.y);
  atomicAdd(op + 2, alpha * hv.z);
  atomicAdd(op + 3, alpha * hv.w);
}

// ---------------------------------------------------------------------------
extern "C" void kernel_launch(void* const* d_in, const int* in_sizes, int n_in,
                              void* d_out, int out_size, void* d_ws, size_t ws_size,
                              hipStream_t stream) {
  const float* x = (const float*)d_in[0];
  const int* ei = (const int*)d_in[1];
  const float* W = (const float*)d_in[2];
  const float* att_src = (const float*)d_in[3];
  const float* att_dst = (const float*)d_in[4];
  const float* bias = (const float*)d_in[5];
  float* out = (float*)d_out;

  const int N = in_sizes[0] / F;
  const int E = in_sizes[1] / 2;
  const int EN = E + N;
  const int* srcA = ei;
  const int* dstA = ei + E;

  float* ws = (float*)d_ws;
  float* hbuf = ws;                              // N*128
  float* asrc = hbuf + (size_t)N * F;            // N*8
  float* adst = asrc + (size_t)N * HEADS;        // N*8
  float* mbuf = adst + (size_t)N * HEADS;        // N*8 (uint then float)
  float* dbuf = mbuf + (size_t)N * HEADS;        // N*8

  {
    long long tot = (long long)N * F;
    int blocks = (int)((tot + 255) / 256);
    gat_init<<<blocks, 256, 0, stream>>>(out, bias, (unsigned*)mbuf, dbuf, N);
  }
  gat_gemm_wmma<<<(N + 15) / 16, 256, 0, stream>>>(x, W, att_src, att_dst,
                                                   hbuf, asrc, adst, N);
  gat_logits_max<<<(EN + 255) / 256, 256, 0, stream>>>(srcA, dstA, asrc, adst,
                                                       (unsigned*)mbuf, E, EN);
  gat_finalize_m<<<((N * HEADS) + 255) / 256, 256, 0, stream>>>((unsigned*)mbuf,
                                                                N * HEADS);
  gat_expsum<<<(EN + 255) / 256, 256, 0, stream>>>(srcA, dstA, asrc, adst,
                                                   mbuf, dbuf, E, EN);
  gat_aggregate<<<(EN + 7) / 8, 256, 0, stream>>>(srcA, dstA, asrc, adst,
                                                  mbuf, dbuf, hbuf, out, E, EN);
}